// BlazeFace_27513560498527
// MI455X (gfx1250) — compile-verified
//
#include <hip/hip_runtime.h>
#include <stdint.h>

// BlazeFace decode + weighted-NMS for MI455X (gfx1250).
// One wave32 per batch element; detections live in LDS (SoA float4 groups);
// anchors staged via CDNA5 async global->LDS DMA; wave-synchronous NMS scan
// with shuffle-butterfly reductions and bitmask early exit.

#define N_ANCH     896
#define APL        28          // anchors per lane = 896/32
#define MAX_DET    256
#define BATCHN     512
#define MIN_SCORE  0.75f
#define MIN_SUPP   0.3f
#define INV_SCALE  0.0078125f  // 1/128 exact
#define SCORE_CLIP 100.0f

typedef int v4i_t __attribute__((vector_size(16)));   // matches builtin param type

__device__ __forceinline__ void wait_async0() {
#if __has_builtin(__builtin_amdgcn_s_wait_asynccnt)
  __builtin_amdgcn_s_wait_asynccnt(0);
#else
  asm volatile("s_wait_asynccnt 0" ::: "memory");
#endif
}

// Async DMA one 16B chunk global -> LDS (ASYNCcnt path, CDNA5).
__device__ __forceinline__ void async_copy_b128(const void* g, void* l) {
#if __has_builtin(__builtin_amdgcn_global_load_async_to_lds_b128)
  __builtin_amdgcn_global_load_async_to_lds_b128(
      (__attribute__((address_space(1))) v4i_t*)(uintptr_t)g,
      (__attribute__((address_space(3))) v4i_t*)(uint32_t)(uintptr_t)l,
      0, 0);
#else
  unsigned ldsOff = (unsigned)(uintptr_t)l;            // generic LDS addr[31:0] == LDS offset
  unsigned long long ga = (unsigned long long)(uintptr_t)g;
  asm volatile("global_load_async_to_lds_b128 %0, %1, off"
               :: "v"(ldsOff), "v"(ga) : "memory");
#endif
}

__global__ __launch_bounds__(32)
void blazeface_nms_kernel(const float* __restrict__ raw_box,
                          const float* __restrict__ raw_score,
                          const float* __restrict__ anchors,
                          float* __restrict__ out) {
  // det coords 0..15 as 4 float4 groups (SoA) + score; ~75 KB total LDS.
  __shared__ float4 sDet4[4][N_ANCH];
  __shared__ float  sScore[N_ANCH];
  __shared__ float4 sAnch[N_ANCH];

  const int b    = blockIdx.x;
  const int lane = threadIdx.x;

  // ---- Stage anchors (shared by all) via async global->LDS DMA ----
  const float4* a4 = (const float4*)anchors;
#pragma unroll 4
  for (int i = 0; i < APL; ++i) {
    int a = (i << 5) + lane;
    async_copy_b128(a4 + a, &sAnch[a]);
  }

  const float4* rb4 = (const float4*)raw_box + (size_t)b * N_ANCH * 4;
  const float*  sc  = raw_score + (size_t)b * N_ANCH;
  __builtin_prefetch(&rb4[lane * 4], 0, 0);  // global_prefetch_b8 warm-up

  wait_async0();
  __syncthreads();

  // ---- Decode 896 anchors into LDS; build per-lane validity bitmask ----
  unsigned remaining = 0u;
  for (int i = 0; i < APL; ++i) {
    int a = (i << 5) + lane;                       // lane owns anchors lane+32*i
    float4 g0 = rb4[a * 4 + 0];
    float4 g1 = rb4[a * 4 + 1];
    float4 g2 = rb4[a * 4 + 2];
    float4 g3 = rb4[a * 4 + 3];
    float4 an = sAnch[a];
    float ax = an.x, ay = an.y, aw = an.z, ah = an.w;

    float xc = g0.x * INV_SCALE * aw + ax;
    float yc = g0.y * INV_SCALE * ah + ay;
    float w  = g0.z * INV_SCALE * aw;
    float h  = g0.w * INV_SCALE * ah;
    float4 d0 = { yc - 0.5f * h, xc - 0.5f * w, yc + 0.5f * h, xc + 0.5f * w };
    float4 d1 = { g1.x * INV_SCALE * aw + ax, g1.y * INV_SCALE * ah + ay,
                  g1.z * INV_SCALE * aw + ax, g1.w * INV_SCALE * ah + ay };
    float4 d2 = { g2.x * INV_SCALE * aw + ax, g2.y * INV_SCALE * ah + ay,
                  g2.z * INV_SCALE * aw + ax, g2.w * INV_SCALE * ah + ay };
    float4 d3 = { g3.x * INV_SCALE * aw + ax, g3.y * INV_SCALE * ah + ay,
                  g3.z * INV_SCALE * aw + ax, g3.w * INV_SCALE * ah + ay };
    sDet4[0][a] = d0; sDet4[1][a] = d1; sDet4[2][a] = d2; sDet4[3][a] = d3;

    float s = sc[a];
    s = fminf(fmaxf(s, -SCORE_CLIP), SCORE_CLIP);
    float sig = 1.0f / (1.0f + __expf(-s));
    sScore[a] = sig;
    remaining |= (sig >= MIN_SCORE ? 1u : 0u) << i;
  }
  __syncthreads();

  float* outRows = out + (size_t)b * MAX_DET * 17;
  float* flags   = out + (size_t)BATCHN * MAX_DET * 17 + (size_t)b * MAX_DET;

  // ---- Wave-synchronous weighted-NMS scan ----
  int step = 0;
  for (; step < MAX_DET; ++step) {
    // argmax over remaining (first-max tie-break like jnp.argmax)
    float bs = -1.0f;            // sigmoid scores are in (0,1)
    int   bi = 0x7fffffff;
    unsigned m = remaining;
    while (m) {
      int i = __builtin_ctz(m); m &= m - 1;
      int a = (i << 5) + lane;
      float s = sScore[a];
      if (s > bs) { bs = s; bi = a; }
    }
#pragma unroll
    for (int off = 16; off > 0; off >>= 1) {
      float os = __shfl_xor(bs, off, 32);
      int   oi = __shfl_xor(bi, off, 32);
      if (os > bs || (os == bs && oi < bi)) { bs = os; bi = oi; }
    }
    if (bi == 0x7fffffff) break;  // remaining empty -> reference emits zeros forever

    float4 rb = sDet4[0][bi];     // LDS broadcast (uniform address)
    float areaA = (rb.z - rb.x) * (rb.w - rb.y);

    float acc[16];
#pragma unroll
    for (int j = 0; j < 16; ++j) acc[j] = 0.f;
    float tot = 0.f, cnt = 0.f;
    unsigned supp = 0u;

    m = remaining;
    while (m) {
      int i = __builtin_ctz(m); m &= m - 1;
      int a = (i << 5) + lane;
      float4 bb = sDet4[0][a];
      float i0 = fmaxf(rb.x, bb.x);
      float i1 = fmaxf(rb.y, bb.y);
      float i2 = fminf(rb.z, bb.z);
      float i3 = fminf(rb.w, bb.w);
      float inter = (i2 - i0) * (i3 - i1);
      float areaB = (bb.z - bb.x) * (bb.w - bb.y);
      float iou   = fmaxf(0.f, inter / (areaA + areaB - inter));
      if (iou > MIN_SUPP) {
        supp |= 1u << i;
        float w = sScore[a];
        tot += w; cnt += 1.f;
        acc[0] += w * bb.x; acc[1] += w * bb.y; acc[2] += w * bb.z; acc[3] += w * bb.w;
        float4 q1 = sDet4[1][a];
        acc[4] += w * q1.x; acc[5] += w * q1.y; acc[6] += w * q1.z; acc[7] += w * q1.w;
        float4 q2 = sDet4[2][a];
        acc[8] += w * q2.x; acc[9] += w * q2.y; acc[10] += w * q2.z; acc[11] += w * q2.w;
        float4 q3 = sDet4[3][a];
        acc[12] += w * q3.x; acc[13] += w * q3.y; acc[14] += w * q3.z; acc[15] += w * q3.w;
      }
    }
    remaining &= ~supp;

    // 18-value butterfly reduction across the wave (all lanes get totals)
#pragma unroll
    for (int off = 16; off > 0; off >>= 1) {
      tot += __shfl_xor(tot, off, 32);
      cnt += __shfl_xor(cnt, off, 32);
#pragma unroll
      for (int j = 0; j < 16; ++j) acc[j] += __shfl_xor(acc[j], off, 32);
    }

    if (lane == 0) {
      float* row = outRows + step * 17;
      bool  many    = cnt > 1.5f;                 // n > 1
      float safe    = (tot > 0.f) ? tot : 1.f;
      float invSafe = 1.f / safe;
      float wscore  = tot / fmaxf(cnt, 1.f);
      if (many) {
#pragma unroll
        for (int j = 0; j < 16; ++j) row[j] = acc[j] * invSafe;
        row[16] = wscore;
      } else {
#pragma unroll
        for (int g = 0; g < 4; ++g) {
          float4 q = sDet4[g][bi];
          row[4 * g + 0] = q.x; row[4 * g + 1] = q.y;
          row[4 * g + 2] = q.z; row[4 * g + 3] = q.w;
        }
        row[16] = sScore[bi];
      }
      flags[step] = 1.0f;
    }
  }

  // Bulk zero-fill for steps after remaining emptied (matches reference exactly)
  for (int idx = step * 17 + lane; idx < MAX_DET * 17; idx += 32)
    outRows[idx] = 0.f;
  for (int idx = step + lane; idx < MAX_DET; idx += 32)
    flags[idx] = 0.f;
}

extern "C" void kernel_launch(void* const* d_in, const int* in_sizes, int n_in,
                              void* d_out, int out_size, void* d_ws, size_t ws_size,
                              hipStream_t stream) {
  const float* raw_box   = (const float*)d_in[0];  // (512, 896, 16) f32
  const float* raw_score = (const float*)d_in[1];  // (512, 896, 1)  f32
  const float* anchors   = (const float*)d_in[2];  // (896, 4)       f32
  float* out = (float*)d_out;                      // outs (512,256,17) ++ flags (512,256)
  (void)in_sizes; (void)n_in; (void)out_size; (void)d_ws; (void)ws_size;

  blazeface_nms_kernel<<<BATCHN, 32, 0, stream>>>(raw_box, raw_score, anchors, out);
}